// Net_83794811945331
// MI455X (gfx1250) — compile-verified
//
#include <hip/hip_runtime.h>
#include <math.h>

// Problem constants (from reference): B=8192, P=64, IN=3, H=20, OUT=10, K=4
#define NBATCH 8192
#define NP     64
#define NH     20
#define NOUT   10
#define NK     4
#define NEG_INF (-__builtin_huge_valf())

// LDS strides chosen so 16-lane column accesses hit 16 distinct banks:
// gcd(stride,64)=4 and stride/4 odd -> {stride*r mod 64 : r=0..15} all distinct.
#define ESTR 44   // edge-feature matrix stride (floats)
#define MSTR 36   // m1 matrix stride (floats)

typedef __attribute__((ext_vector_type(2))) float v2f;
typedef __attribute__((ext_vector_type(8))) float v8f;

__device__ __forceinline__ float eluf(float x) { return x > 0.f ? x : expm1f(x); }

// order-preserving float <-> unsigned (for LDS atomic max incl. -inf)
__device__ __forceinline__ unsigned fflip(float f) {
    unsigned u = __float_as_uint(f);
    return (u & 0x80000000u) ? ~u : (u | 0x80000000u);
}
__device__ __forceinline__ float funflip(unsigned u) {
    return (u & 0x80000000u) ? __uint_as_float(u & 0x7fffffffu) : __uint_as_float(~u);
}

__global__ __launch_bounds__(64) void dgcnn_kernel(
    const float* __restrict__ x,
    const float* __restrict__ Wi0, const float* __restrict__ bi0,
    const float* __restrict__ Wi1, const float* __restrict__ bi1,
    const float* __restrict__ Wi2, const float* __restrict__ bi2,
    const float* __restrict__ We0a, const float* __restrict__ be0a,
    const float* __restrict__ We0b, const float* __restrict__ be0b,
    const float* __restrict__ We1a, const float* __restrict__ be1a,
    const float* __restrict__ We1b, const float* __restrict__ be1b,
    const float* __restrict__ Wo0, const float* __restrict__ bo0,
    const float* __restrict__ Wo1, const float* __restrict__ bo1,
    const float* __restrict__ Wo2, const float* __restrict__ bo2,
    float* __restrict__ out)
{
    __shared__ float    s_h[NP][NH];          // current features (f32, exact)
    __shared__ float    s_hn[NP][NH];         // edgeconv output
    __shared__ float    s_score[NP][NP + 1];  // stride 65 -> conflict-free row scans
    __shared__ float    s_sq[NP];
    __shared__ int      s_valid[NP];
    __shared__ int      s_idx[NP][NK];
    __shared__ float    s_E[128][ESTR];       // 128 edge rows x 40 features (per chunk)
    __shared__ float    s_M1[128][MSTR];      // GEMM1 output (N padded to 32)
    __shared__ float    s_vj[128];            // per-edge neighbor-validity
    __shared__ float    s_W1[40][32];         // W (40x20) zero-padded to 32 cols
    __shared__ float    s_b1[32];
    __shared__ float    s_W2[20][32];
    __shared__ float    s_b2[32];
    __shared__ unsigned s_pool[NP + 1][NH];
    __shared__ int      s_cnt[NP + 1];
    __shared__ unsigned s_gbits[NH];
    __shared__ float    s_vec[NH];
    __shared__ float    s_vec2[NH];

    const int b    = blockIdx.x;
    const int tid  = threadIdx.x;   // 0..63 == point id
    const int lane = tid & 31;
    const int wave = tid >> 5;
    const int half = lane >> 4;     // WMMA lane-half selects K pair
    const int row  = lane & 15;     // WMMA M (A) / N (B,D) index

    // ================= Phase 1: input MLP (per point) =================
    {
        float* wbuf  = &s_W1[0][0];   // flat staging (>=60)
        float* wbuf2 = &s_W2[0][0];   // flat staging (>=400)
        for (int i = tid; i < 3 * NH; i += 64) wbuf[i] = Wi0[i];
        for (int i = tid; i < NH * NH; i += 64) wbuf2[i] = Wi1[i];
        if (tid < NH) { s_b1[tid] = bi0[tid]; s_b2[tid] = bi1[tid]; }
        __syncthreads();

        float xin[3];
#pragma unroll
        for (int d = 0; d < 3; ++d) xin[d] = x[(size_t)b * NP * 3 + tid * 3 + d];

        float h0[NH], h1[NH];
#pragma unroll
        for (int j = 0; j < NH; ++j) {
            float a = s_b1[j];
#pragma unroll
            for (int d = 0; d < 3; ++d) a += xin[d] * wbuf[d * NH + j];
            h0[j] = eluf(a);
        }
#pragma unroll
        for (int j = 0; j < NH; ++j) {
            float a = s_b2[j];
#pragma unroll
            for (int d = 0; d < NH; ++d) a += h0[d] * wbuf2[d * NH + j];
            h1[j] = eluf(a);
        }
        __syncthreads();
        for (int i = tid; i < NH * NH; i += 64) wbuf2[i] = Wi2[i];
        if (tid < NH) s_b2[tid] = bi2[tid];
        __syncthreads();

        float sq = 0.f;
#pragma unroll
        for (int j = 0; j < NH; ++j) {
            float a = s_b2[j];
#pragma unroll
            for (int d = 0; d < NH; ++d) a += h1[d] * wbuf2[d * NH + j];
            const float v = eluf(a);
            s_h[tid][j] = v;
            sq += v * v;
        }
        s_sq[tid]    = sq;
        s_valid[tid] = 1;
    }
    __syncthreads();

    // ================= 2x { kNN -> EdgeConv -> Pool } =================
    int sel0 = 0;   // idx[...,0] of this thread's point (for pooling)
    for (int round = 0; round < 2; ++round) {
        const float* Wap = round ? We1a : We0a;
        const float* bap = round ? be1a : be0a;
        const float* Wbp = round ? We1b : We0b;
        const float* bbp = round ? be1b : be0b;

        // ---- Gram matrix h.h^T via exact f32 WMMA (K=20 = 5 steps of 4) ----
        for (int t8 = 0; t8 < 8; ++t8) {
            const int tile = wave * 8 + t8;
            const int ti = tile >> 2, tj = tile & 3;
            const int pi = ti * 16 + row;   // A row (M = lane%16)
            const int pj = tj * 16 + row;   // B col (N = lane%16)
            v8f c = {};
#pragma unroll
            for (int ks = 0; ks < 5; ++ks) {
                const int kk = ks * 4 + half * 2;   // lane-half selects K pair
                v2f a, bf;
                a[0]  = s_h[pi][kk];
                a[1]  = s_h[pi][kk + 1];
                bf[0] = s_h[pj][kk];
                bf[1] = s_h[pj][kk + 1];
                c = __builtin_amdgcn_wmma_f32_16x16x4_f32(
                        false, a, false, bf, (short)0, c, false, false);
            }
            // Branchless epilogue: all LDS reads unconditional, masks via selects.
            const int   gj  = tj * 16 + row;
            const float sqj = s_sq[gj];
            const int   vld = s_valid[gj];
            const int   gb  = ti * 16 + half * 8;
            float sqi[8];
#pragma unroll
            for (int r = 0; r < 8; ++r) sqi[r] = s_sq[gb + r];
#pragma unroll
            for (int r = 0; r < 8; ++r) {
                const int gi = gb + r;
                float s = 2.f * c[r] - sqi[r] - sqj;
                const int bad = (gi == gj) | (vld == 0);
                s_score[gi][gj] = bad ? NEG_INF : s;
            }
        }
        __syncthreads();

        // ---- stable top-K=4 per row (matches jax.lax.top_k ties) ----
        int sel[NK];
#pragma unroll
        for (int k = 0; k < NK; ++k) {
            float best = NEG_INF;
            int   bj = 0;
            bool  found = false;
            for (int j = 0; j < NP; ++j) {
                bool taken = false;
#pragma unroll
                for (int q = 0; q < NK; ++q)
                    if (q < k) taken |= (sel[q] == j);
                if (taken) continue;
                const float s = s_score[tid][j];
                if (!found || s > best) { best = s; bj = j; found = true; }
            }
            sel[k] = bj;
            s_idx[tid][k] = bj;
        }
        sel0 = sel[0];

        // ---- stage EdgeConv weights, zero-padded N 20->32 ----
        for (int i = tid; i < 40 * 32; i += 64) {
            const int r = i >> 5, c = i & 31;
            s_W1[r][c] = (c < NH) ? Wap[r * NH + c] : 0.f;
        }
        for (int i = tid; i < 20 * 32; i += 64) {
            const int r = i >> 5, c = i & 31;
            s_W2[r][c] = (c < NH) ? Wbp[r * NH + c] : 0.f;
        }
        if (tid < 32) {
            s_b1[tid] = (tid < NH) ? bap[tid] : 0.f;
            s_b2[tid] = (tid < NH) ? bbp[tid] : 0.f;
        }
        __syncthreads();   // covers s_idx + weights

        // ---- EdgeConv as two dense GEMMs on WMMA, 2 chunks of 128 edge rows ----
        for (int chunk = 0; chunk < 2; ++chunk) {
            // build edge matrix E[rr][0:20]=x_i, E[rr][20:40]=x_j-x_i (2 rows/thread)
#pragma unroll
            for (int rr2 = 0; rr2 < 2; ++rr2) {
                const int rr = tid * 2 + rr2;              // 0..127
                const int p  = chunk * 32 + (rr >> 2);
                const int k  = rr & 3;
                const int j  = s_idx[p][k];
                s_vj[rr] = s_valid[j] ? 1.f : 0.f;
#pragma unroll
                for (int d = 0; d < NH; ++d) {
                    const float hp = s_h[p][d];
                    s_E[rr][d]      = hp;
                    s_E[rr][NH + d] = s_h[j][d] - hp;
                }
            }
            __syncthreads();

            for (int mt = wave; mt < 8; mt += 2) {   // 4 M-tiles per wave
                // GEMM1: E(16x40) x W1(40x32), K = 10 steps of 4, exact
#pragma unroll
                for (int nt = 0; nt < 2; ++nt) {
                    v8f acc = {};
#pragma unroll
                    for (int ks = 0; ks < 10; ++ks) {
                        const int kk = ks * 4 + half * 2;
                        v2f a, bf;
                        a[0]  = s_E[mt * 16 + row][kk];
                        a[1]  = s_E[mt * 16 + row][kk + 1];
                        bf[0] = s_W1[kk][nt * 16 + row];
                        bf[1] = s_W1[kk + 1][nt * 16 + row];
                        acc = __builtin_amdgcn_wmma_f32_16x16x4_f32(
                                false, a, false, bf, (short)0, acc, false, false);
                    }
#pragma unroll
                    for (int r = 0; r < 8; ++r) {
                        const int rr = mt * 16 + half * 8 + r;
                        const int n  = nt * 16 + row;
                        s_M1[rr][n] = eluf(acc[r] + s_b1[n]);
                    }
                }
                // GEMM2: M1(16x20) x W2(20x32), K = 5 steps of 4, exact.
                // Same wave wrote these M1 rows -> only dscnt wait needed (compiler).
#pragma unroll
                for (int nt = 0; nt < 2; ++nt) {
                    v8f acc = {};
#pragma unroll
                    for (int ks = 0; ks < 5; ++ks) {
                        const int kk = ks * 4 + half * 2;
                        v2f a, bf;
                        a[0]  = s_M1[mt * 16 + row][kk];
                        a[1]  = s_M1[mt * 16 + row][kk + 1];
                        bf[0] = s_W2[kk][nt * 16 + row];
                        bf[1] = s_W2[kk + 1][nt * 16 + row];
                        acc = __builtin_amdgcn_wmma_f32_16x16x4_f32(
                                false, a, false, bf, (short)0, acc, false, false);
                    }
                    // Epilogue: each lane holds 2 complete points x 4 edges of col n.
                    const int n = nt * 16 + row;
                    if (n < NH) {
                        const int base = mt * 16 + half * 8;
                        float s0 = 0.f, s1 = 0.f;
#pragma unroll
                        for (int r = 0; r < 4; ++r)
                            s0 += eluf(acc[r] + s_b2[n]) * s_vj[base + r];
#pragma unroll
                        for (int r = 4; r < 8; ++r)
                            s1 += eluf(acc[r] + s_b2[n]) * s_vj[base + r];
                        const int p0 = chunk * 32 + mt * 4 + half * 2;
                        s_hn[p0][n]     = s0;
                        s_hn[p0 + 1][n] = s1;
                    }
                }
            }
            __syncthreads();   // s_E reused next chunk; s_hn complete after last
        }

        // ---- Pool: segment max into cluster = min(i, nn0) ----
        for (int i = tid; i < (NP + 1) * NH; i += 64) (&s_pool[0][0])[i] = fflip(NEG_INF);
        for (int i = tid; i < NP + 1; i += 64) s_cnt[i] = 0;
        __syncthreads();
        const int myvalid = s_valid[tid];
        {
            const int nn0 = sel0;
            const int cl  = tid < nn0 ? tid : nn0;
            const int c   = myvalid ? cl : NP;
#pragma unroll
            for (int f = 0; f < NH; ++f) {
                const float v = myvalid ? s_hn[tid][f] : NEG_INF;
                atomicMax(&s_pool[c][f], fflip(v));   // ds_max_rtn_u32
            }
            atomicAdd(&s_cnt[c], myvalid);
        }
        __syncthreads();
        {
            const int nv = s_cnt[tid] > 0;
            float sq = 0.f;
#pragma unroll
            for (int f = 0; f < NH; ++f) {
                const float v = nv ? funflip(s_pool[tid][f]) : 0.f;
                s_h[tid][f] = v;
                sq += v * v;
            }
            s_sq[tid]    = sq;
            s_valid[tid] = nv;
        }
        __syncthreads();
    }

    // ================= global max over valid points =================
    if (tid < NH) s_gbits[tid] = fflip(NEG_INF);
    __syncthreads();
    {
        const int myvalid = s_valid[tid];
#pragma unroll
        for (int f = 0; f < NH; ++f) {
            const float v = myvalid ? s_h[tid][f] : NEG_INF;
            atomicMax(&s_gbits[f], fflip(v));
        }
    }
    __syncthreads();

    // ================= head MLP + log_softmax =================
    if (tid < NH) {
        float a = bo0[tid];
        for (int f = 0; f < NH; ++f) a += funflip(s_gbits[f]) * Wo0[f * NH + tid];
        s_vec[tid] = eluf(a);
    }
    __syncthreads();
    if (tid < NH) {
        float a = bo1[tid];
        for (int f = 0; f < NH; ++f) a += s_vec[f] * Wo1[f * NH + tid];
        s_vec2[tid] = eluf(a);
    }
    __syncthreads();
    if (tid < NOUT) {
        float a = bo2[tid];
        for (int f = 0; f < NH; ++f) a += s_vec2[f] * Wo2[f * NOUT + tid];
        s_vec[tid] = a;   // logits
    }
    __syncthreads();
    if (tid < NOUT) {
        float m = NEG_INF;
        for (int j = 0; j < NOUT; ++j) m = fmaxf(m, s_vec[j]);
        float se = 0.f;
        for (int j = 0; j < NOUT; ++j) se += expf(s_vec[j] - m);
        out[(size_t)b * NOUT + tid] = s_vec[tid] - m - logf(se);
    }
}

extern "C" void kernel_launch(void* const* d_in, const int* in_sizes, int n_in,
                              void* d_out, int out_size, void* d_ws, size_t ws_size,
                              hipStream_t stream) {
    const float* x    = (const float*)d_in[0];
    const float* Wi0  = (const float*)d_in[1];
    const float* bi0  = (const float*)d_in[2];
    const float* Wi1  = (const float*)d_in[3];
    const float* bi1  = (const float*)d_in[4];
    const float* Wi2  = (const float*)d_in[5];
    const float* bi2  = (const float*)d_in[6];
    const float* We0a = (const float*)d_in[7];
    const float* be0a = (const float*)d_in[8];
    const float* We0b = (const float*)d_in[9];
    const float* be0b = (const float*)d_in[10];
    const float* We1a = (const float*)d_in[11];
    const float* be1a = (const float*)d_in[12];
    const float* We1b = (const float*)d_in[13];
    const float* be1b = (const float*)d_in[14];
    const float* Wo0  = (const float*)d_in[15];
    const float* bo0  = (const float*)d_in[16];
    const float* Wo1  = (const float*)d_in[17];
    const float* bo1  = (const float*)d_in[18];
    const float* Wo2  = (const float*)d_in[19];
    const float* bo2  = (const float*)d_in[20];
    float* out = (float*)d_out;

    dgcnn_kernel<<<NBATCH, 64, 0, stream>>>(
        x, Wi0, bi0, Wi1, bi1, Wi2, bi2,
        We0a, be0a, We0b, be0b, We1a, be1a, We1b, be1b,
        Wo0, bo0, Wo1, bo1, Wo2, bo2, out);
}